// GeneralAffinityCalculator_59725815218716
// MI455X (gfx1250) — compile-verified
//
#include <hip/hip_runtime.h>
#include <hip/hip_bf16.h>

#define LATENT 128
#define KQ     32
#define BB     2
#define NN     65536
#define KK     32

typedef float v2f __attribute__((ext_vector_type(2)));
typedef float v8f __attribute__((ext_vector_type(8)));

// ---------------------------------------------------------------------------
// Phase 1: fused K/Q projection via fp32 WMMA.
//   ks[row, d] = features[row, :] . Wk[d, :] + bk[d]
//   qs[row, d] = features[row, :] . Wq[d, :] + bq[d]
// One workgroup (4 wave32s) per 16-row block; each wave owns one 16x16 tile:
//   wave0: K cols 0-15, wave1: K cols 16-31, wave2: Q cols 0-15, wave3: Q 16-31
// V_WMMA_F32_16X16X4_F32 accumulated over K=128 in steps of 4.
// ---------------------------------------------------------------------------
__global__ __launch_bounds__(128) void affinity_proj_wmma(
    const float* __restrict__ features,
    const float* __restrict__ Wk, const float* __restrict__ bk,
    const float* __restrict__ Wq, const float* __restrict__ bq,
    float* __restrict__ ks, float* __restrict__ qs)
{
    const int lane = threadIdx.x & 31;
    const int wave = threadIdx.x >> 5;         // 0..3
    const int rowBase = blockIdx.x << 4;       // 16 rows per block
    const int colBase = (wave & 1) << 4;       // 0 or 16
    const bool isQ = (wave >> 1) != 0;

    const float* __restrict__ W    = isQ ? Wq : Wk;
    const float* __restrict__ bias = isQ ? bq : bk;
    float* __restrict__ out        = isQ ? qs : ks;

    // ISA 16x4 fp32 A layout: lanes 0-15 -> M=lane, K={kk+0,kk+1};
    //                         lanes 16-31 -> M=lane-16, K={kk+2,kk+3}.
    // B (=W^T, 4x16) mirrors this with M -> N.
    const int m    = lane & 15;
    const int koff = (lane >> 4) << 1;         // 0 or 2

    const float* arow = features + (size_t)(rowBase + m) * LATENT + koff;
    const float* brow = W        + (size_t)(colBase + m) * LATENT + koff;

    v8f c = {};
#pragma unroll 4
    for (int kk = 0; kk < LATENT; kk += 4) {
        v2f a = *(const v2f*)(arow + kk);      // global_load_b64
        v2f b = *(const v2f*)(brow + kk);      // global_load_b64 (L2-resident)
        // 8 args: (neg_a, A, neg_b, B, c_mod, C, reuse_a, reuse_b)
        c = __builtin_amdgcn_wmma_f32_16x16x4_f32(
                false, a, false, b, (short)0, c, false, false);
    }

    // D layout: VGPR r, lanes 0-15 -> M=r; lanes 16-31 -> M=r+8; N=lane&15.
    const float bv = bias[colBase + m];
    const int rbase = rowBase + ((lane >> 4) << 3);
#pragma unroll
    for (int r = 0; r < 8; ++r)
        out[(size_t)(rbase + r) * KQ + colBase + m] = c[r] + bv;
}

// ---------------------------------------------------------------------------
// Phase 2: gather + batched dot. One thread per (b, n, k) pair.
// Both 16 MB tables are L2-resident (192 MB L2), so the random row gathers
// never touch HBM; HBM traffic is just the index streams + output.
// ---------------------------------------------------------------------------
__global__ __launch_bounds__(256) void affinity_gather_dot(
    const int* __restrict__ x_idx, const int* __restrict__ y_idx,
    const float* __restrict__ ks, const float* __restrict__ qs,
    float* __restrict__ out)
{
    const long long p = (long long)blockIdx.x * 256 + threadIdx.x;
    const int b = (int)(p / ((long long)NN * KK));   // N*K = 2^21
    const int x = x_idx[p];
    const int y = y_idx[p];

    const float4* __restrict__ xr =
        (const float4*)(ks + (((size_t)b * NN + (size_t)x) << 5));
    const float4* __restrict__ yr =
        (const float4*)(qs + (((size_t)b * NN + (size_t)y) << 5));

    float s = 0.f;
#pragma unroll
    for (int i = 0; i < 8; ++i) {
        float4 a = xr[i];                      // global_load_b128
        float4 q = yr[i];                      // global_load_b128
        s = fmaf(a.x, q.x, s);
        s = fmaf(a.y, q.y, s);
        s = fmaf(a.z, q.z, s);
        s = fmaf(a.w, q.w, s);
    }
    out[p] = s * 0.17677669529663687f;         // 32^-0.5
}

// ---------------------------------------------------------------------------
extern "C" void kernel_launch(void* const* d_in, const int* in_sizes, int n_in,
                              void* d_out, int out_size, void* d_ws, size_t ws_size,
                              hipStream_t stream)
{
    const int*   indices  = (const int*)d_in[0];   // [3, B, N, K]
    const float* features = (const float*)d_in[1]; // [B, N, 128]
    const float* Wk       = (const float*)d_in[2]; // [32, 128]
    const float* bk       = (const float*)d_in[3]; // [32]
    const float* Wq       = (const float*)d_in[4]; // [32, 128]
    const float* bq       = (const float*)d_in[5]; // [32]
    float* out = (float*)d_out;                    // [B, N, K]

    const long long BNK = (long long)BB * NN * KK; // 4,194,304

    float* ks = (float*)d_ws;                      // 16 MB
    float* qs = ks + (size_t)BB * NN * KQ;         // 16 MB

    // Phase 1: 131072 rows / 16 = 8192 workgroups x 128 threads (4 full waves).
    affinity_proj_wmma<<<(BB * NN) / 16, 128, 0, stream>>>(
        features, Wk, bk, Wq, bq, ks, qs);

    // Phase 2: one thread per pair; rows 1 and 2 of indices are x/y ids.
    affinity_gather_dot<<<(int)(BNK / 256), 256, 0, stream>>>(
        indices + BNK, indices + 2 * BNK, ks, qs, out);
}